// MeshUpConv_89489938579756
// MI455X (gfx1250) — compile-verified
//
#include <hip/hip_runtime.h>

typedef __attribute__((ext_vector_type(2))) float v2f;
typedef __attribute__((ext_vector_type(8))) float v8f;

#define BPITCH 80   // padded LDS row pitch (floats); 80 mod 64 = 16 -> halves of the
                    // wave hit disjoint bank ranges on ds_load (conflict-free)

// ---------------------------------------------------------------------------
// WMMA f32 GEMM:  C[m, s*64 + o] (+= bias[o]) = sum_k A[m,k] * B[s][k][o]
//   A: [M, lda] row-major, lda = K
//   B: nslots consecutive [K,64] row-major panels; s is BLOCK-uniform
//   C: row stride ldc, slot s writes columns [s*64, s*64+64)
// Each block: stages the 32KB (K=128) B panel into LDS once (coalesced
// float4), then 8 waves each compute a 16x64 tile (4 WMMA accumulators),
// looping K in steps of 4 with B fragments served from LDS.
// v_wmma_f32_16x16x4_f32 fragment layouts per CDNA5 ISA 7.12.2:
//   A 16x4 : lane l holds A[row0 + l%16][kb], A[..][kb+1], kb = k + 2*(l/16)
//   B 4x16 : lane l holds B[kb][n], B[kb+1][n],            n  = l%16
//   C 16x16: vgpr r, lane l -> M = row0 + r + 8*(l/16), N = l%16
// ---------------------------------------------------------------------------
__global__ __launch_bounds__(256) void gemm_wmma_f32(
    const float* __restrict__ A, int lda,
    const float* __restrict__ B,
    float* __restrict__ C, int ldc,
    const float* __restrict__ bias,
    int M, int K, int nslots, int mblocks)
{
    __shared__ float bsh[128 * BPITCH];          // 40KB max (K<=128)

    const int s    = blockIdx.x / mblocks;       // block-uniform slot
    const int mblk = blockIdx.x - s * mblocks;
    const int wvid = threadIdx.x >> 5;
    const int lane = threadIdx.x & 31;
    const int mtiles = (M + 15) >> 4;
    const int mtile  = mblk * 8 + wvid;

    // ---- stage B panel into LDS, coalesced float4, padded pitch ----
    const float* __restrict__ Bs = B + (size_t)s * K * 64;
    const int nElem = K * 64;
    for (int g = threadIdx.x * 4; g < nElem; g += 256 * 4) {
        const float4 v = *(const float4*)(Bs + g);      // float4 never crosses a row
        const int row = g >> 6, col = g & 63;
        *(float4*)(&bsh[row * BPITCH + col]) = v;
    }
    __syncthreads();

    if (mtile < mtiles) {                        // wave-uniform: EXEC all-1s at WMMA
        const int row0 = mtile << 4;
        const int half = lane >> 4;              // 0 or 1
        const int l16  = lane & 15;

        int arow = row0 + l16; if (arow > M - 1) arow = M - 1;   // clamp loads
        const float* __restrict__ Abase = A + (size_t)arow * lda;

        v8f acc0 = {}, acc1 = {}, acc2 = {}, acc3 = {};

        for (int k = 0; k < K; k += 4) {
            const int kb = k + (half << 1);
            v2f a;
            a.x = Abase[kb];
            a.y = Abase[kb + 1];

            const float* __restrict__ bp0 = &bsh[kb * BPITCH];   // row kb
            const float* __restrict__ bp1 = bp0 + BPITCH;        // row kb+1
            v2f b0, b1, b2, b3;
            b0.x = bp0[l16];      b0.y = bp1[l16];
            b1.x = bp0[16 + l16]; b1.y = bp1[16 + l16];
            b2.x = bp0[32 + l16]; b2.y = bp1[32 + l16];
            b3.x = bp0[48 + l16]; b3.y = bp1[48 + l16];

            acc0 = __builtin_amdgcn_wmma_f32_16x16x4_f32(false, a, false, b0, (short)0, acc0, false, false);
            acc1 = __builtin_amdgcn_wmma_f32_16x16x4_f32(false, a, false, b1, (short)0, acc1, false, false);
            acc2 = __builtin_amdgcn_wmma_f32_16x16x4_f32(false, a, false, b2, (short)0, acc2, false, false);
            acc3 = __builtin_amdgcn_wmma_f32_16x16x4_f32(false, a, false, b3, (short)0, acc3, false, false);
        }

        const int colbase = s * 64;
        float bv0 = 0.f, bv1 = 0.f, bv2 = 0.f, bv3 = 0.f;
        if (bias) {
            bv0 = bias[l16];      bv1 = bias[16 + l16];
            bv2 = bias[32 + l16]; bv3 = bias[48 + l16];
        }
#pragma unroll
        for (int r = 0; r < 8; ++r) {
            const int m = row0 + r + (half << 3);
            if (m < M) {
                float* __restrict__ crow = C + (size_t)m * ldc + colbase;
                crow[l16]      = acc0[r] + bv0;
                crow[16 + l16] = acc1[r] + bv1;
                crow[32 + l16] = acc2[r] + bv2;
                crow[48 + l16] = acc3[r] + bv3;
            }
        }
    }
}

// ---------------------------------------------------------------------------
// Edge gather + B-spline weighting + scatter-add (memory-bound phase).
// One wave32 per edge; lane owns channels {2*lane, 2*lane+1} (float2 gathers,
// 256B coalesced per slot).  out[dst, o] += sum_s b0[k0]*b1[k1] * xW[src, s, o]
// ---------------------------------------------------------------------------
__global__ __launch_bounds__(256) void edge_scatter(
    const float* __restrict__ xW,          // [N, 576]
    const int*   __restrict__ ei,          // [2, E]
    const float* __restrict__ ea,          // [E, 2]
    float* __restrict__ out, int ostride, int E)
{
    const int e    = (blockIdx.x * blockDim.x + threadIdx.x) >> 5;
    const int lane = threadIdx.x & 31;
    if (e >= E) return;

    const int src = ei[e];
    const int dst = ei[E + e];
    const float u0 = ea[2 * e];
    const float u1 = ea[2 * e + 1];

    float w0[3], w1[3];
    w0[0] = 0.5f * (1.f - u0) * (1.f - u0);
    w0[1] = -u0 * u0 + u0 + 0.5f;
    w0[2] = 0.5f * u0 * u0;
    w1[0] = 0.5f * (1.f - u1) * (1.f - u1);
    w1[1] = -u1 * u1 + u1 + 0.5f;
    w1[2] = 0.5f * u1 * u1;

    const float* __restrict__ xr = xW + (size_t)src * 576 + 2 * lane;
    float a0 = 0.f, a1 = 0.f;
#pragma unroll
    for (int k1 = 0; k1 < 3; ++k1) {
#pragma unroll
        for (int k0 = 0; k0 < 3; ++k0) {
            const float c = w0[k0] * w1[k1];
            const int ss = k0 + 3 * k1;
            const float2 v = *(const float2*)(xr + ss * 64);   // 8B aligned
            a0 += c * v.x;
            a1 += c * v.y;
        }
    }
    float* __restrict__ op = out + (size_t)dst * ostride + 2 * lane;
    __hip_atomic_fetch_add(op,     a0, __ATOMIC_RELAXED, __HIP_MEMORY_SCOPE_AGENT);
    __hip_atomic_fetch_add(op + 1, a1, __ATOMIC_RELAXED, __HIP_MEMORY_SCOPE_AGENT);
}

// relu cols [0,64) of a stride-128 buffer in place, and copy skip into cols [64,128)
__global__ __launch_bounds__(256) void relu_concat(
    float* __restrict__ h, const float* __restrict__ skip, int N64)
{
    const int i = blockIdx.x * blockDim.x + threadIdx.x;
    if (i >= N64) return;
    const int row = i >> 6, col = i & 63;
    float* __restrict__ hr = h + (size_t)row * 128;
    const float v = hr[col];
    hr[col] = v > 0.f ? v : 0.f;
    hr[64 + col] = skip[i];
}

__global__ __launch_bounds__(256) void relu_inplace(float* __restrict__ p, int n)
{
    const int i = blockIdx.x * blockDim.x + threadIdx.x;
    if (i >= n) return;
    const float v = p[i];
    p[i] = v > 0.f ? v : 0.f;
}

extern "C" void kernel_launch(void* const* d_in, const int* in_sizes, int n_in,
                              void* d_out, int out_size, void* d_ws, size_t ws_size,
                              hipStream_t stream)
{
    const float* x     = (const float*)d_in[0];   // [N,128]
    const float* skip  = (const float*)d_in[1];   // [N,64]
    const int*   ei    = (const int*)  d_in[2];   // [2,E]
    const float* ea    = (const float*)d_in[3];   // [E,2]
    const float* W1    = (const float*)d_in[4];   // [9,128,64]
    const float* root1 = (const float*)d_in[5];   // [128,64]
    const float* b1    = (const float*)d_in[6];   // [64]
    const float* W2    = (const float*)d_in[7];   // [9,64,64]
    const float* root2 = (const float*)d_in[8];   // [64,64]
    const float* b2    = (const float*)d_in[9];   // [64]
    float* out = (float*)d_out;

    const int N = in_sizes[0] / 128;
    const int E = in_sizes[3] / 2;

    char* ws = (char*)d_ws;
    float* xWb  = (float*)ws;                                               // [N,576]
    float* hbuf = (float*)(ws + (size_t)N * 576 * 4);                       // [N,128]
    float* h2   = (float*)(ws + (size_t)N * 576 * 4 + (size_t)N * 128 * 4); // [N,64]

    const int mtiles  = (N + 15) / 16;
    const int mblocks = (mtiles + 7) / 8;        // 8 waves (m-tiles) per block
    const int eblocks  = (E + 7) / 8;
    const int n64      = N * 64;
    const int epblocks = (n64 + 255) / 256;

    // ---- Layer 1: h1 = relu(conv1(x)); hbuf = concat(h1, skip) ----
    gemm_wmma_f32<<<9 * mblocks, 256, 0, stream>>>(x, 128, W1, xWb, 576, nullptr, N, 128, 9, mblocks);
    gemm_wmma_f32<<<1 * mblocks, 256, 0, stream>>>(x, 128, root1, hbuf, 128, b1, N, 128, 1, mblocks);
    edge_scatter<<<eblocks, 256, 0, stream>>>(xWb, ei, ea, hbuf, 128, E);
    relu_concat<<<epblocks, 256, 0, stream>>>(hbuf, skip, n64);

    // ---- Layer 2: h2 = relu(conv1(hbuf)) ----
    gemm_wmma_f32<<<9 * mblocks, 256, 0, stream>>>(hbuf, 128, W1, xWb, 576, nullptr, N, 128, 9, mblocks);
    gemm_wmma_f32<<<1 * mblocks, 256, 0, stream>>>(hbuf, 128, root1, h2, 64, b1, N, 128, 1, mblocks);
    edge_scatter<<<eblocks, 256, 0, stream>>>(xWb, ei, ea, h2, 64, E);
    relu_inplace<<<epblocks, 256, 0, stream>>>(h2, n64);

    // ---- Layer 3: out = relu(conv2(h2)) ----
    gemm_wmma_f32<<<9 * mblocks, 256, 0, stream>>>(h2, 64, W2, xWb, 576, nullptr, N, 64, 9, mblocks);
    gemm_wmma_f32<<<1 * mblocks, 256, 0, stream>>>(h2, 64, root2, out, 64, b2, N, 64, 1, mblocks);
    edge_scatter<<<eblocks, 256, 0, stream>>>(xWb, ei, ea, out, 64, E);
    relu_inplace<<<epblocks, 256, 0, stream>>>(out, n64);
}